// WARPLoss_67302137528417
// MI455X (gfx1250) — compile-verified
//
#include <hip/hip_runtime.h>
#include <hip/hip_bf16.h>

#define BATCH 16384
#define NLAB  4096
#define TMAX  64
#define WAVES_PER_BLOCK 8
#define CAND_CAP 512

typedef __attribute__((ext_vector_type(2))) float v2f;
typedef __attribute__((ext_vector_type(8))) float v8f;

// ---------- deterministic stateless noise: 24-bit uniform in [0, 2^24) ----------
__device__ __forceinline__ unsigned int mix24(unsigned int a, unsigned int b) {
  unsigned int s = a * 0x9E3779B9u + b * 0x85EBCA6Bu + 0x6A09E667u;
  s ^= s >> 16; s *= 0x7FEB352Du;
  s ^= s >> 15; s *= 0x846CA68Bu;
  s ^= s >> 16;
  return s >> 8;
}

// wave-local LDS ordering: CDNA5 split dependency counter wait
__device__ __forceinline__ void wave_sync_lds() {
  __builtin_amdgcn_wave_barrier();
  asm volatile("s_wait_dscnt 0x0" ::: "memory");
  __builtin_amdgcn_wave_barrier();
}

__device__ __forceinline__ int wave_max_i(int v) {
  #pragma unroll
  for (int off = 16; off; off >>= 1) {
    int o = __shfl_xor(v, off, 32);
    v = o > v ? o : v;
  }
  return v;
}

__device__ __forceinline__ unsigned long long wave_max_u64(unsigned long long v) {
  #pragma unroll
  for (int off = 16; off; off >>= 1) {
    unsigned long long o = __shfl_xor(v, off, 32);
    v = o > v ? o : v;
  }
  return v;
}

__device__ __forceinline__ int wave_sum_i(int v) {
  #pragma unroll
  for (int off = 16; off; off >>= 1) v += __shfl_xor(v, off, 32);
  return v;
}

// ---------------------------------------------------------------------------
// Kernel 1: one wave32 per batch row. Single streaming pass over targets and
// inputs (the only HBM traffic); noise is regenerated in ALU for the rank
// computations. Writes per-row loss into d_ws.
// ---------------------------------------------------------------------------
__global__ __launch_bounds__(32 * WAVES_PER_BLOCK) void warp_loss_rows(
    const float* __restrict__ inputs, const float* __restrict__ targets,
    float* __restrict__ row_loss) {
  __shared__ unsigned long long cand[WAVES_PER_BLOCK][CAND_CAP];
  __shared__ unsigned int       ccnt[WAVES_PER_BLOCK];
  __shared__ unsigned long long sel[WAVES_PER_BLOCK];

  const int lane = threadIdx.x & 31;
  const int wv   = threadIdx.x >> 5;
  const int b    = blockIdx.x * WAVES_PER_BLOCK + wv;

  const float* inp = inputs  + (size_t)b * NLAB;
  const float* tgt = targets + (size_t)b * NLAB;
  __builtin_prefetch(inp, 0, 1);
  __builtin_prefetch(tgt, 0, 1);

  // ---- pass A: positive index from the one-hot targets row ----
  int posi = 0;
  const float4* t4 = (const float4*)tgt;
  for (int i = lane; i < NLAB / 4; i += 32) {
    float4 t = t4[i];
    int j = 4 * i;
    if (t.x > 0.5f) posi = j;
    if (t.y > 0.5f) posi = j + 1;
    if (t.z > 0.5f) posi = j + 2;
    if (t.w > 0.5f) posi = j + 3;
  }
  posi = wave_max_i(posi);
  const float pos_score = inp[posi];
  const float thresh    = pos_score - 1.0f;   // margin > 0  <=>  x > thresh

  // ---- pass B: stream inputs once; best (max-noise) hitting negative ----
  unsigned long long bestKey = 0ull;
  const float4* x4 = (const float4*)inp;
  for (int i = lane; i < NLAB / 4; i += 32) {
    float4 x = x4[i];
    int j = 4 * i;
    float xv[4] = {x.x, x.y, x.z, x.w};
    #pragma unroll
    for (int k = 0; k < 4; ++k) {
      int jj = j + k;
      if (jj == posi) continue;                 // positive sorts last (noise=-1)
      if (xv[k] > thresh) {
        unsigned long long key =
            ((unsigned long long)mix24((unsigned)b, (unsigned)jj) << 32) |
            (unsigned)jj;
        if (key > bestKey) bestKey = key;
      }
    }
  }
  bestKey = wave_max_u64(bestKey);
  const bool hit = (bestKey != 0ull);
  const unsigned int ubstar = (unsigned int)(bestKey >> 32);

  // ---- pass C (ALU only): rank of best hit = #negatives with larger noise ----
  int r = 0;
  if (hit) {
    int cnt = 0;
    for (int j = lane; j < NLAB; j += 32) {
      if (j == posi) continue;
      cnt += (mix24((unsigned)b, (unsigned)j) > ubstar) ? 1 : 0;
    }
    r = wave_sum_i(cnt);
  }
  const bool found = hit && (r < TMAX);   // first hit falls inside the T-window

  int  neg_idx;
  int  num_trials;
  bool set_flag;

  if (found) {
    neg_idx    = (int)(bestKey & 0xffffffffu);
    num_trials = r + 1;
    set_flag   = true;
  } else {
    // Need the rank-(T-1) noise element (the T-th random negative).
    // Threshold + gather candidates to LDS, then O(c^2) rank search.
    unsigned int tthr = 16441672u;            // ~0.98 * 2^24  (E[c] ~ 82)
    int c = 0;
    #pragma unroll 1
    for (int pass = 0; pass < 2; ++pass) {
      int cnt = 0;
      for (int j = lane; j < NLAB; j += 32) {
        if (j == posi) continue;
        cnt += (mix24((unsigned)b, (unsigned)j) > tthr) ? 1 : 0;
      }
      c = wave_sum_i(cnt);
      if (c >= TMAX) break;
      tthr = 15099494u;                       // ~0.90 * 2^24  (E[c] ~ 410)
    }
    if (lane == 0) ccnt[wv] = 0u;
    wave_sync_lds();
    for (int j = lane; j < NLAB; j += 32) {
      if (j == posi) continue;
      unsigned int ub = mix24((unsigned)b, (unsigned)j);
      if (ub > tthr) {
        unsigned int slot = atomicAdd(&ccnt[wv], 1u);
        if (slot < CAND_CAP)
          cand[wv][slot] = ((unsigned long long)ub << 32) | (unsigned)j;
      }
    }
    wave_sync_lds();
    unsigned int cc = ccnt[wv];
    if (cc > CAND_CAP) cc = CAND_CAP;
    if (lane == 0) sel[wv] = cand[wv][0];
    wave_sync_lds();
    for (unsigned int k = lane; k < cc; k += 32) {
      unsigned long long key = cand[wv][k];
      int cg = 0;
      for (unsigned int m = 0; m < cc; ++m) cg += (cand[wv][m] > key) ? 1 : 0;
      if (cg == TMAX - 1) sel[wv] = key;      // exactly one winner (keys unique)
    }
    wave_sync_lds();
    unsigned long long skey = sel[wv];
    neg_idx    = (int)(skey & 0xffffffffu);
    num_trials = TMAX;
    set_flag   = (inp[neg_idx] >= thresh);    // last margin >= 0
  }

  if (lane == 0) {
    float loss = 0.0f;
    if (set_flag) {
      float lw = logf(floorf((float)(NLAB - 1) / (float)num_trials));
      loss = lw * (1.0f - pos_score + inp[neg_idx]);
    }
    row_loss[b] = loss;
  }
}

// ---------------------------------------------------------------------------
// Kernel 2: batch reduction via chained V_WMMA_F32_16X16X4_F32.
// A = 16x4 ones, B carries 64 loss values per issue, C accumulates:
// D[m,n] = sum_k B[k,n], so summing row 0 of D over the 16 columns yields the
// total of every value fed through B. One wave, EXEC all ones.
// ---------------------------------------------------------------------------
__global__ __launch_bounds__(32) void warp_sum_wmma(
    const float* __restrict__ vals, float* __restrict__ out, int n) {
  const int lane = threadIdx.x;
  v2f a; a[0] = 1.0f; a[1] = 1.0f;
  v8f c = {};
  for (int base = 0; base < n; base += 64) {
    v2f bm;
    bm[0] = vals[base + lane];
    bm[1] = vals[base + 32 + lane];
    c = __builtin_amdgcn_wmma_f32_16x16x4_f32(
        /*neg_a=*/false, a, /*neg_b=*/false, bm,
        /*c_mod=*/(short)0, c, /*reuse_a=*/false, /*reuse_b=*/false);
  }
  // lanes 0..15 of VGPR0 hold D[0, N=lane]; lanes 16..31 duplicate the columns
  float v = (lane < 16) ? c[0] : 0.0f;
  #pragma unroll
  for (int off = 16; off; off >>= 1) v += __shfl_xor(v, off, 32);
  if (lane == 0) out[0] = v;
}

extern "C" void kernel_launch(void* const* d_in, const int* in_sizes, int n_in,
                              void* d_out, int out_size, void* d_ws, size_t ws_size,
                              hipStream_t stream) {
  const float* inputs  = (const float*)d_in[0];
  const float* targets = (const float*)d_in[1];
  float* out      = (float*)d_out;
  float* row_loss = (float*)d_ws;     // BATCH floats of scratch

  warp_loss_rows<<<BATCH / WAVES_PER_BLOCK, 32 * WAVES_PER_BLOCK, 0, stream>>>(
      inputs, targets, row_loss);
  warp_sum_wmma<<<1, 32, 0, stream>>>(row_loss, out, BATCH);
}